// SpatialNumDualDescriptorAB2_36790689858202
// MI455X (gfx1250) — compile-verified
//
#include <hip/hip_runtime.h>

typedef __attribute__((ext_vector_type(2))) float v2f;
typedef __attribute__((ext_vector_type(8))) float v8f;

#define MD    256   // feature dim m
#define L12   49    // L1*L2 basis count
#define CPAD  50    // 49 basis rows + 1 zero row (for clamped B-fragment indexing)
#define HN    63    // H - RANK1 + 1
#define WN    63    // W - RANK2 + 1

// ---------------------------------------------------------------------------
// Kernel 1: tildeB[c][m] = sum_n Bbasis[c][n] * M[n][m]   (row 49 == 0)
// ---------------------------------------------------------------------------
__global__ void __launch_bounds__(256) make_tildeB_kernel(
    const float* __restrict__ Mmat,      // [256,256] row-major M[n][m]
    const float* __restrict__ Bbasis,    // [49,256]
    float* __restrict__ tildeB)          // [50,256]
{
    int idx = blockIdx.x * blockDim.x + threadIdx.x;  // [0, 50*256)
    int c = idx >> 8;
    int m = idx & 255;
    float acc = 0.0f;
    if (c < L12) {
        const float* brow = Bbasis + c * MD;
        for (int n = 0; n < MD; ++n)
            acc = fmaf(brow[n], Mmat[n * MD + m], acc);
    }
    tildeB[idx] = acc;
}

// ---------------------------------------------------------------------------
// Kernel 2: fused skinny GEMM (WMMA f32 16x16x4) + separable-weight reduction.
// Block = 256 threads = 8 waves; block owns 2 image rows (2*64 pixels);
// each wave owns 16 consecutive pixels of one row (A-tile rows = pixels).
// D[p, c] = arr[p,:] . tildeB[c,:]  for c in 0..63 (cols 49..63 are zero).
// weight(h,w,c) = wv(h, c/7) * wv(w, c%7)  (window-membership count factors).
// ---------------------------------------------------------------------------
__global__ void __launch_bounds__(256) fused_gemm_reduce_kernel(
    const float* __restrict__ arr,       // [32*64*64, 256]
    const float* __restrict__ tildeB,    // [50, 256]
    float* __restrict__ Tglob)           // [32, 49], pre-zeroed
{
    __shared__ float sB[CPAD * MD];      // 50 KB
    __shared__ float T_local[L12];

    const int tid = threadIdx.x;

    // Stage tildeB into LDS; zero the per-block accumulators.
    for (int i = tid; i < CPAD * MD; i += 256) sB[i] = tildeB[i];
    if (tid < L12) T_local[tid] = 0.0f;
    __syncthreads();

    const int wave  = tid >> 5;                      // 0..7
    const int lane  = tid & 31;
    const int row   = blockIdx.x * 2 + (wave >> 2);  // global (b*64 + h), 0..2047
    const int b     = row >> 6;
    const int h     = row & 63;
    const int wbase = (wave & 3) * 16;               // first pixel column of tile

    const int col   = lane & 15;          // A-row (pixel) / B-col (c) within tile
    const int khalf = (lane >> 4) * 2;    // K sub-offset: 0 or 2

    // A fragment source: 16 consecutive pixels of this image row.
    const float* aptr = arr + ((size_t)(row * 64 + wbase + col)) * MD + khalf;

    // B fragment sources for the 4 c-tiles (clamped to the zero row for c>=49).
    const float* bp0 = &sB[min(0 * 16 + col, 49) * MD + khalf];
    const float* bp1 = &sB[min(1 * 16 + col, 49) * MD + khalf];
    const float* bp2 = &sB[min(2 * 16 + col, 49) * MD + khalf];
    const float* bp3 = &sB[min(3 * 16 + col, 49) * MD + khalf];

    v8f acc[4] = {v8f{}, v8f{}, v8f{}, v8f{}};

    // K loop: 256 / 4 = 64 steps; 1 global b64 load, 4 LDS b64 loads, 4 WMMAs.
    for (int k0 = 0; k0 < MD; k0 += 4) {
        v2f a  = *(const v2f*)(aptr + k0);
        v2f f0 = *(const v2f*)(bp0 + k0);
        v2f f1 = *(const v2f*)(bp1 + k0);
        v2f f2 = *(const v2f*)(bp2 + k0);
        v2f f3 = *(const v2f*)(bp3 + k0);
        acc[0] = __builtin_amdgcn_wmma_f32_16x16x4_f32(false, a, false, f0,
                                                       (short)0, acc[0], false, false);
        acc[1] = __builtin_amdgcn_wmma_f32_16x16x4_f32(false, a, false, f1,
                                                       (short)0, acc[1], false, false);
        acc[2] = __builtin_amdgcn_wmma_f32_16x16x4_f32(false, a, false, f2,
                                                       (short)0, acc[2], false, false);
        acc[3] = __builtin_amdgcn_wmma_f32_16x16x4_f32(false, a, false, f3,
                                                       (short)0, acc[3], false, false);
    }

    // ---- Separable-weight scatter --------------------------------------
    // C/D layout: lane holds column n = lane&15 (-> c); element e is pixel
    // row M = 8*(lane>>4) + e (-> w = woff + e).
    // wv(x,a) = [x<=62 && x%7==a] + [x>=1 && (x-1)%7==a]
    const int woff = wbase + ((lane >> 4) << 3);

    int   w7[8], w7m[8];
    float v0[8], v1[8];
#pragma unroll
    for (int e = 0; e < 8; ++e) {
        int w  = woff + e;
        w7[e]  = w % 7;
        w7m[e] = (w + 6) % 7;                 // (w-1) mod 7; w==0 killed by v1
        v0[e]  = (w <= WN - 1) ? 1.0f : 0.0f; // w <= 62
        v1[e]  = (w >= 1)      ? 1.0f : 0.0f;
    }
    const int   h7  = h % 7;
    const int   h7m = (h + 6) % 7;
    const float hv0 = (h <= HN - 1) ? 1.0f : 0.0f;
    const float hv1 = (h >= 1)      ? 1.0f : 0.0f;

#pragma unroll
    for (int t = 0; t < 4; ++t) {
        int c = t * 16 + col;
        if (c < L12) {
            int c1 = c / 7;
            int c2 = c % 7;
            float wh = (h7 == c1 ? hv0 : 0.0f) + (h7m == c1 ? hv1 : 0.0f);
            if (wh != 0.0f) {
                float part = 0.0f;
#pragma unroll
                for (int e = 0; e < 8; ++e) {
                    float ww = (w7[e] == c2 ? v0[e] : 0.0f) +
                               (w7m[e] == c2 ? v1[e] : 0.0f);
                    part = fmaf(ww, acc[t][e], part);
                }
                if (part != 0.0f) atomicAdd(&T_local[c], wh * part); // ds_add_f32
            }
        }
    }
    __syncthreads();
    if (tid < L12) atomicAdd(&Tglob[b * L12 + tid], T_local[tid]);
}

// ---------------------------------------------------------------------------
// Kernel 3: rep[b,n] = (1/(4*63*63)) * sum_c T[b,c] * Acoeff[n,c]
// ---------------------------------------------------------------------------
__global__ void __launch_bounds__(256) finalize_kernel(
    const float* __restrict__ Tglob,     // [32, 49]
    const float* __restrict__ Acoeff,    // [256, 49]
    float* __restrict__ out)             // [32, 256]
{
    __shared__ float sT[L12];
    const int b = blockIdx.x;
    const int n = threadIdx.x;
    if (n < L12) sT[n] = Tglob[b * L12 + n];
    __syncthreads();
    const float scale = 1.0f / (4.0f * 63.0f * 63.0f);
    float acc = 0.0f;
#pragma unroll
    for (int c = 0; c < L12; ++c)
        acc = fmaf(sT[c], Acoeff[n * L12 + c], acc);
    out[b * MD + n] = acc * scale;
}

// ---------------------------------------------------------------------------
extern "C" void kernel_launch(void* const* d_in, const int* in_sizes, int n_in,
                              void* d_out, int out_size, void* d_ws, size_t ws_size,
                              hipStream_t stream) {
    const float* arr    = (const float*)d_in[0];  // [32,64,64,256]
    const float* Mmat   = (const float*)d_in[1];  // [256,256]
    const float* Acoeff = (const float*)d_in[2];  // [256,49]
    const float* Bbasis = (const float*)d_in[3];  // [49,256]
    float* out = (float*)d_out;                   // [32,256]

    float* tildeB = (float*)d_ws;                  // 50*256 floats = 51200 B
    float* Tglob  = (float*)d_ws + CPAD * MD;      // 32*49 floats

    hipMemsetAsync(Tglob, 0, 32 * L12 * sizeof(float), stream);
    make_tildeB_kernel<<<CPAD, 256, 0, stream>>>(Mmat, Bbasis, tildeB);
    fused_gemm_reduce_kernel<<<1024, 256, 0, stream>>>(arr, tildeB, Tglob);
    finalize_kernel<<<32, 256, 0, stream>>>(Tglob, Acoeff, out);
}